// Grid_10737418240653
// MI455X (gfx1250) — compile-verified
//
#include <hip/hip_runtime.h>
#include <math.h>
#include <stdint.h>

#define TABLE_SIZE 524288u
#define TABLE_MASK (TABLE_SIZE - 1u)
#define PRIME_Y 2654435761u
#define PRIME_Z 805459861u
#define TILE 256

__device__ __forceinline__ float2 lerp2(float2 a, float2 b, float w, float u) {
    float2 r;
    r.x = a.x * u + b.x * w;
    r.y = a.y * u + b.y * w;
    return r;
}

// Trilinear hash-grid interpolation for one point.
__device__ __forceinline__ float2 interp_point(float xs, float ys, float zs,
                                               const float2* __restrict__ tbl) {
    float fx = floorf(xs), fy = floorf(ys), fz = floorf(zs);
    float cx = ceilf(xs),  cy = ceilf(ys),  cz = ceilf(zs);
    float w0 = xs - fx, w1 = ys - fy, w2 = zs - fz;

    // Coordinates are non-negative, so the reference's int64 hash reduces
    // exactly to uint32 arithmetic (XOR/mask depend only on low bits).
    unsigned hx0 = (unsigned)(int)fx;
    unsigned hx1 = (unsigned)(int)cx;
    unsigned hy0 = (unsigned)(int)fy * PRIME_Y;
    unsigned hy1 = (unsigned)(int)cy * PRIME_Y;
    unsigned hz0 = (unsigned)(int)fz * PRIME_Z;
    unsigned hz1 = (unsigned)(int)cz * PRIME_Z;

    // Corner k = (x<<2)|(y<<1)|z, matching OFFSETS in the reference.
    unsigned idx[8];
    idx[0] = (hx0 ^ hy0 ^ hz0) & TABLE_MASK;
    idx[1] = (hx0 ^ hy0 ^ hz1) & TABLE_MASK;
    idx[2] = (hx0 ^ hy1 ^ hz0) & TABLE_MASK;
    idx[3] = (hx0 ^ hy1 ^ hz1) & TABLE_MASK;
    idx[4] = (hx1 ^ hy0 ^ hz0) & TABLE_MASK;
    idx[5] = (hx1 ^ hy0 ^ hz1) & TABLE_MASK;
    idx[6] = (hx1 ^ hy1 ^ hz0) & TABLE_MASK;
    idx[7] = (hx1 ^ hy1 ^ hz1) & TABLE_MASK;

    // Issue all 8 gathers (global_load_b64) before any use -> latency overlap.
    float2 v[8];
#pragma unroll
    for (int k = 0; k < 8; ++k) v[k] = tbl[idx[k]];

    float u0 = 1.0f - w0, u1 = 1.0f - w1, u2 = 1.0f - w2;
    float2 p00 = lerp2(v[0], v[4], w0, u0);
    float2 p01 = lerp2(v[1], v[5], w0, u0);
    float2 p10 = lerp2(v[2], v[6], w0, u0);
    float2 p11 = lerp2(v[3], v[7], w0, u0);
    float2 p0  = lerp2(p00, p10, w1, u1);
    float2 p1  = lerp2(p01, p11, w1, u1);
    return lerp2(p0, p1, w2, u2);
}

// Main kernel: one block = one tile of 256 points (exact tiles only).
// X tile (256*3 floats = 3072 B) is staged into LDS with the gfx1250
// async-to-LDS path: 192 lanes x global_load_async_to_lds_b128, then
// s_wait_asynccnt + barrier.
__global__ __launch_bounds__(TILE) void hashgrid_tile_kernel(
    const float* __restrict__ X, const float* __restrict__ table,
    float* __restrict__ out, float scale) {
    __shared__ __align__(16) float sx[TILE * 3];

    const unsigned tid  = threadIdx.x;
    const unsigned base = (unsigned)blockIdx.x * TILE;

    // Async-stage this tile's coordinates into LDS.
    const float* src = X + (size_t)base * 3u;
    unsigned lds_base = (unsigned)(uintptr_t)(&sx[0]);  // LDS byte offset (low 32 bits of generic ptr)
    if (tid < (TILE * 3u * 4u) / 16u) {                 // 192 lanes move 16 B each
        unsigned off  = tid * 16u;
        unsigned ldsa = lds_base + off;
        asm volatile("global_load_async_to_lds_b128 %0, %1, %2 offset:0"
                     :
                     : "v"(ldsa), "v"(off), "s"(src)
                     : "memory");
    }
    asm volatile("s_wait_asynccnt 0x0" ::: "memory");
    __syncthreads();

    float xs = sx[3u * tid + 0u] * scale;
    float ys = sx[3u * tid + 1u] * scale;
    float zs = sx[3u * tid + 2u] * scale;

    float2 r = interp_point(xs, ys, zs, (const float2*)table);
    ((float2*)out)[base + tid] = r;  // 8-byte aligned -> global_store_b64
}

// Tail kernel (defensive; N = 2^21 is an exact multiple of 256 so this is
// normally never launched). Direct loads, bounds-guarded.
__global__ __launch_bounds__(TILE) void hashgrid_tail_kernel(
    const float* __restrict__ X, const float* __restrict__ table,
    float* __restrict__ out, float scale, unsigned start, unsigned n) {
    unsigned i = start + blockIdx.x * blockDim.x + threadIdx.x;
    if (i >= n) return;
    float xs = X[3u * i + 0u] * scale;
    float ys = X[3u * i + 1u] * scale;
    float zs = X[3u * i + 2u] * scale;
    float2 r = interp_point(xs, ys, zs, (const float2*)table);
    out[2u * i + 0u] = r.x;
    out[2u * i + 1u] = r.y;
}

extern "C" void kernel_launch(void* const* d_in, const int* in_sizes, int n_in,
                              void* d_out, int out_size, void* d_ws, size_t ws_size,
                              hipStream_t stream) {
    (void)n_in; (void)out_size; (void)d_ws; (void)ws_size;
    const float* X     = (const float*)d_in[0];   // (N, 3) f32
    const float* table = (const float*)d_in[1];   // (T, 2) f32
    float* out         = (float*)d_out;           // (N, 2) f32

    const long long n = (long long)in_sizes[0] / 3;  // number of points

    // scale = RES - 1, computed in double exactly like the NumPy reference,
    // then rounded to f32 (matches jnp's f32 * weak-scalar multiply).
    const double growth = exp((log(512.0) - log(16.0)) / 15.0);
    const double res    = pow(growth, 8.0) * 16.0;
    const float  scale  = (float)(res - 1.0);

    const long long full = n / TILE;
    const long long rem  = n % TILE;

    if (full > 0) {
        hashgrid_tile_kernel<<<(unsigned)full, TILE, 0, stream>>>(X, table, out, scale);
    }
    if (rem > 0) {
        hashgrid_tail_kernel<<<1, TILE, 0, stream>>>(X, table, out, scale,
                                                     (unsigned)(full * TILE), (unsigned)n);
    }
}